// ACSL_83751862272634
// MI455X (gfx1250) — compile-verified
//
#include <hip/hip_runtime.h>

#define N_I   65536
#define N_CLS 1203
#define N_C   1204          // N_CLS + 1 (background column)
#define NV4   301           // N_C / 4 float4s per row (row stride 4816 B, 16B aligned)
#define ROWS_PER_BLOCK 8    // 8 wave32 per 256-thread block, one row per wave
#define NBLOCKS (N_I / ROWS_PER_BLOCK)   // 8192
#define SCORE_LOGIT 0.8472978603872034f  // ln(0.7/0.3): sigmoid(x)>=0.7 <=> x>=this

typedef float vf4 __attribute__((ext_vector_type(4)));
typedef float v2f __attribute__((ext_vector_type(2)));
typedef float v8f __attribute__((ext_vector_type(8)));

__device__ __forceinline__ float softplus_f(float x) {
  // stable: max(x,0) + log1p(exp(-|x|)); arg of log is in [1,2] so plain log is fine
  float e = __expf(-__builtin_fabsf(x));
  return fmaxf(x, 0.0f) + __logf(1.0f + e);
}

__global__ __launch_bounds__(256) void acsl_partial_kernel(
    const float* __restrict__ logits,
    const int*   __restrict__ labels,
    const int*   __restrict__ sel_rand,
    const int*   __restrict__ cat_freq,
    float*       __restrict__ block_sums)
{
  __shared__ unsigned char codes[N_C + 4];   // per-column class code, byte packed
  __shared__ float wave_sums[ROWS_PER_BLOCK];

  // stage column codes once per block: 0=rare 1=common 2=freq 3=background col
  for (int c = threadIdx.x; c < N_C; c += 256)
    codes[c] = (c < N_CLS) ? (unsigned char)cat_freq[c] : (unsigned char)3;
  __syncthreads();

  const int wave = threadIdx.x >> 5;
  const int lane = threadIdx.x & 31;
  const int row  = blockIdx.x * ROWS_PER_BLOCK + wave;

  // row scalars are wave-uniform: force into SGPRs
  const int lab = __builtin_amdgcn_readfirstlane(labels[row]);
  const int sel = __builtin_amdgcn_readfirstlane(sel_rand[row]);

  const vf4* rowp = (const vf4*)(logits + (size_t)row * N_C);

  float acc = 0.0f;

  if (lab != N_CLS) {
    // foreground row: weight = (x >= logit(0.7)) | (c == label); loss = softplus - x*[c==label]
    for (int j = lane; j < NV4; j += 32) {
      vf4 v = __builtin_nontemporal_load(&rowp[j]);
      int c0 = j << 2;
#pragma unroll
      for (int k = 0; k < 4; ++k) {
        float x = v[k];
        bool  t = (c0 + k) == lab;
        bool  keep = t | (x >= SCORE_LOGIT);
        float l = softplus_f(x) - (t ? x : 0.0f);
        acc += keep ? l : 0.0f;
      }
    }
  } else {
    // background row: weight from per-column code + per-row subsampling draw
    const unsigned lut = (sel < 10 ? 1u : 0u) | (sel < 100 ? 2u : 0u) | 4u | 8u;
    const unsigned* codes4 = (const unsigned*)codes;
    for (int j = lane; j < NV4; j += 32) {
      vf4 v = __builtin_nontemporal_load(&rowp[j]);
      unsigned cw = codes4[j];                 // 4 column codes per ds_load_b32
      int c0 = j << 2;
#pragma unroll
      for (int k = 0; k < 4; ++k) {
        float    x    = v[k];
        unsigned code = (cw >> (8 * k)) & 0xffu;
        bool     w    = (lut >> code) & 1u;    // code 3 (bg col) and 2 (freq) always on
        bool     t    = (c0 + k) == lab;       // only true at the bg column here
        float    l    = softplus_f(x) - (t ? x : 0.0f);
        acc += w ? l : 0.0f;
      }
    }
  }

  // ---- wave32 reduction through the matrix pipe ----
  // A = ones(16x4), B = 4x16 with each lane contributing {acc, 0}:
  // D[m][n] = sum_k B[k][n]  => lane n of d[0] holds acc[n&15] + acc[(n&15)+16].
  // EXEC is all-1s here (loop divergence has reconverged; fg/bg branch was wave-uniform).
  v2f a; a[0] = 1.0f; a[1] = 1.0f;
  v2f b; b[0] = acc;  b[1] = 0.0f;
  v8f c = {};
  c = __builtin_amdgcn_wmma_f32_16x16x4_f32(false, a, false, b, (short)0, c, false, false);
  float val = c[0];
#pragma unroll
  for (int off = 8; off >= 1; off >>= 1)     // 16 distinct col-sums left -> 4 xor steps
    val += __shfl_xor(val, off, 32);

  if (lane == 0) wave_sums[wave] = val;
  __syncthreads();
  if (threadIdx.x == 0) {
    float s = 0.0f;
#pragma unroll
    for (int w = 0; w < ROWS_PER_BLOCK; ++w) s += wave_sums[w];
    block_sums[blockIdx.x] = s;
  }
}

__global__ __launch_bounds__(256) void acsl_reduce_kernel(
    const float* __restrict__ block_sums, float* __restrict__ out)
{
  __shared__ float lds[256];
  float s = 0.0f;
  for (int i = threadIdx.x; i < NBLOCKS; i += 256) s += block_sums[i];
  lds[threadIdx.x] = s;
  __syncthreads();
  for (int off = 128; off >= 1; off >>= 1) {
    if ((int)threadIdx.x < off) lds[threadIdx.x] += lds[threadIdx.x + off];
    __syncthreads();
  }
  if (threadIdx.x == 0) out[0] = lds[0] * (1.0f / (float)N_I);
}

extern "C" void kernel_launch(void* const* d_in, const int* in_sizes, int n_in,
                              void* d_out, int out_size, void* d_ws, size_t ws_size,
                              hipStream_t stream) {
  const float* cls_logits = (const float*)d_in[0];
  const int*   labels     = (const int*)d_in[1];
  const int*   sel_rand   = (const int*)d_in[2];
  const int*   cat_freq   = (const int*)d_in[3];
  float* block_sums = (float*)d_ws;            // 8192 floats = 32 KB scratch
  float* out        = (float*)d_out;

  acsl_partial_kernel<<<NBLOCKS, 256, 0, stream>>>(cls_logits, labels, sel_rand,
                                                   cat_freq, block_sums);
  acsl_reduce_kernel<<<1, 256, 0, stream>>>(block_sums, out);
}